// STICKYKVCache_LayerWise_75239237091857
// MI455X (gfx1250) — compile-verified
//
#include <hip/hip_runtime.h>

// ---- problem constants (fixed at module construction in the reference) ----
#define NH     16      // heads
#define SEQ    4096    // q_len == kv_len
#define SINK   4
#define OMEGA  64
#define MAXW   2048    // window_scores / local_history width
#define NWIN   59      // eviction-candidate windows: (3840-4)//64
#define TWIN   63      // full-review windows:       (4092)//64
#define KPB    128     // keys per block (8 waves x 16 keys)

typedef float v2f __attribute__((ext_vector_type(2)));
typedef float v8f __attribute__((ext_vector_type(8)));

// Kernel A: obs[h][k] = sum_q attn[0,h,q,k]  via f32 WMMA (ones-matrix trick).
// grid.x = NH * (SEQ/KPB); 256 threads = 8 waves; each wave owns 16 key cols.
__global__ __launch_bounds__(256)
void sticky_colsum_wmma(const float* __restrict__ attn, float* __restrict__ obs) {
    const int tid  = threadIdx.x;
    const int wave = tid >> 5;
    const int lane = tid & 31;
    const int blk  = blockIdx.x;
    const int h    = blk / (SEQ / KPB);
    const int kg   = blk % (SEQ / KPB);
    const int kbase = kg * KPB + wave * 16;
    const int halfr = lane >> 4;      // 0 or 1: selects query row within fragment
    const int ncol  = lane & 15;      // key column within the 16-wide tile

    // pointer to attn[h][halfr][kbase+ncol]; rows advance by SEQ elements
    const float* p = attn + (size_t)h * SEQ * SEQ + (size_t)halfr * SEQ + (kbase + ncol);

    v2f a; a[0] = 1.0f; a[1] = 1.0f;    // A = ones(16x4): row mapping irrelevant
    v8f c0 = {}; v8f c1 = {};           // two accumulators to break D->C chain

    #pragma unroll 2
    for (int q = 0; q < SEQ; q += 8) {
        v2f b0, b1;
        b0[0] = p[(size_t)(q + 0) * SEQ];   // rows {q..q+3} across the fragment
        b0[1] = p[(size_t)(q + 2) * SEQ];
        b1[0] = p[(size_t)(q + 4) * SEQ];   // rows {q+4..q+7}
        b1[1] = p[(size_t)(q + 6) * SEQ];
        // D[m][n] = sum_k B[k][n] + C[m][n]   (A == ones)
        c0 = __builtin_amdgcn_wmma_f32_16x16x4_f32(false, a, false, b0,
                                                   (short)0, c0, false, false);
        c1 = __builtin_amdgcn_wmma_f32_16x16x4_f32(false, a, false, b1,
                                                   (short)0, c1, false, false);
    }

    // D VGPR0, lanes 0..15 hold (M=0, N=lane): the column sums for this tile.
    if (lane < 16) {
        obs[(size_t)h * SEQ + kbase + lane] = c0[0] + c1[0];
    }
}

// Kernel B: window reductions + scatter. One thread per (head, window slot).
// grid = (MAXW/256, NH).
__global__ __launch_bounds__(256)
void sticky_window_scatter(const float* __restrict__ obs,
                           const float* __restrict__ ws_in,
                           float* __restrict__ ws_out,
                           float* __restrict__ lh_out) {
    const int h = blockIdx.y;
    const int w = blockIdx.x * blockDim.x + threadIdx.x;   // 0..MAXW-1

    float sum = 0.0f;
    if (w < TWIN) {                     // windows 0..58 shared with win_scores
        const float* p = obs + (size_t)h * SEQ + SINK + (size_t)w * OMEGA;
        #pragma unroll 8
        for (int i = 0; i < OMEGA; ++i) sum += p[i];
    }

    const size_t wsi = ((size_t)h * MAXW + w) * 3;
    if (w < NWIN) {
        ws_out[wsi + 0] = sum;
        ws_out[wsi + 1] = (float)w;
        ws_out[wsi + 2] = (float)w;
    } else {                            // pass through NaN-initialized buffer
        ws_out[wsi + 0] = ws_in[wsi + 0];
        ws_out[wsi + 1] = ws_in[wsi + 1];
        ws_out[wsi + 2] = ws_in[wsi + 2];
    }
    lh_out[(size_t)h * MAXW + w] = (w < TWIN) ? sum : 0.0f;
}

extern "C" void kernel_launch(void* const* d_in, const int* in_sizes, int n_in,
                              void* d_out, int out_size, void* d_ws, size_t ws_size,
                              hipStream_t stream) {
    const float* attn  = (const float*)d_in[0];   // [1,16,4096,4096] f32
    const float* ws_in = (const float*)d_in[1];   // [16,2048,3] f32 (NaN)
    // d_in[2] (local_history) is zeroed by the reference; output lh tail is 0.

    float* obs    = (float*)d_ws;                          // 16*4096 f32 scratch
    float* ws_out = (float*)d_out;                         // 16*2048*3
    float* lh_out = ws_out + (size_t)NH * MAXW * 3;        // 16*2048

    sticky_colsum_wmma<<<NH * (SEQ / KPB), 256, 0, stream>>>(attn, obs);
    sticky_window_scatter<<<dim3(MAXW / 256, NH), 256, 0, stream>>>(obs, ws_in,
                                                                   ws_out, lh_out);
}